// VarlenNSA_65369402245677
// MI455X (gfx1250) — compile-verified
//
#include <hip/hip_runtime.h>

typedef float v2f __attribute__((ext_vector_type(2)));
typedef float v8f __attribute__((ext_vector_type(8)));

#define NSA_SEQ   2048
#define NSA_KV    2
#define NSA_H     4
#define NSA_QH    8
#define NSA_DIM   128
#define NSA_NBLK  32
#define NSA_SCALE 0.08838834764831845f   // 128^-0.5
#define KV_STRIDE (NSA_KV*NSA_DIM)       // row stride of k/v in floats

static __device__ __forceinline__ v8f wmma_f32(v2f a, v2f b, v8f c) {
  // V_WMMA_F32_16X16X4_F32 : D(16x16,f32) = A(16x4,f32) * B(4x16,f32) + C
  return __builtin_amdgcn_wmma_f32_16x16x4_f32(false, a, false, b, (short)0, c,
                                               false, false);
}

static __device__ __forceinline__ float redmax16(float x) {
  x = fmaxf(x, __shfl_xor(x, 1, 32));
  x = fmaxf(x, __shfl_xor(x, 2, 32));
  x = fmaxf(x, __shfl_xor(x, 4, 32));
  x = fmaxf(x, __shfl_xor(x, 8, 32));
  return x;
}
static __device__ __forceinline__ float redsum16(float x) {
  x += __shfl_xor(x, 1, 32);
  x += __shfl_xor(x, 2, 32);
  x += __shfl_xor(x, 4, 32);
  x += __shfl_xor(x, 8, 32);
  return x;
}

// 16x16 score tile = Q(16x128) * K^T : 32 chained fp32 WMMAs.
// qa: preloaded A-layout (lane holds Q[row=lane%16][4k+2hi .. +1]).
// krow: pointer to this lane's key row (key index = lane%16), contiguous 128 floats? no:
// caller passes per-lane key-row base; element stride 1 in dim.
static __device__ __forceinline__ v8f qk_tile(const v2f* qa,
                                              const float* __restrict__ krow,
                                              int hi) {
  v8f acc = {};
#pragma unroll
  for (int kk = 0; kk < 32; ++kk) {
    v2f b = *(const v2f*)(krow + 4 * kk + 2 * hi);
    acc = wmma_f32(qa[kk], b, acc);
  }
  return acc;
}

// ---------------- kernel 0: zero output ----------------
__global__ void nsa_zero(float* __restrict__ p, int n) {
  int i = blockIdx.x * 256 + threadIdx.x;
  if (i < n) p[i] = 0.0f;
}

// ---------------- kernel 1: block compression (+ zero pooled) ----------------
__global__ void nsa_compress(const float* __restrict__ k, const float* __restrict__ v,
                             const float* __restrict__ w_k, const float* __restrict__ b_k,
                             const float* __restrict__ w_v, const float* __restrict__ b_v,
                             float* __restrict__ kc, float* __restrict__ vc,
                             float* __restrict__ pooled) {
  int tid = blockIdx.x * blockDim.x + threadIdx.x;
  if (tid < 4096) pooled[tid] = 0.0f;
  if (tid >= 32768) return;
  int isV = tid >> 14;
  int r   = tid & 16383;             // ((b*2+kv)*32+n)*128+e
  int e   = r & 127;
  int n   = (r >> 7) & 31;
  int kvi = (r >> 12) & 1;
  int bi  = r >> 13;
  const float* src = isV ? v : k;
  const float* w   = isV ? w_v : w_k;
  float bias = isV ? b_v[0] : b_k[0];
  float acc = 0.0f;
  int rowbase = bi * NSA_SEQ + n * 64;
  for (int t = 0; t < 64; ++t)
    acc += w[t] * src[(size_t)(rowbase + t) * KV_STRIDE + kvi * NSA_DIM + e];
  (isV ? vc : kc)[r] = acc + bias;
}

// ---------------- kernel 2: compressed attention + pooled scores ----------------
__global__ __launch_bounds__(32) void nsa_cmp_attn(
    const float* __restrict__ q, const float* __restrict__ kc,
    const float* __restrict__ vc, const float* __restrict__ wg,
    const float* __restrict__ bg, float* __restrict__ pooled,
    float* __restrict__ out) {
  __shared__ float ldsP[16 * 32];
  __shared__ float ldsG[16];
  const int w = blockIdx.x;
  const int qt = w & 127, h = (w >> 7) & 3, kvi = (w >> 9) & 1, bi = w >> 10;
  const int L = threadIdx.x, lo = L & 15, hi = L >> 4;
  const int q0 = qt * 16;
  const int hoff = kvi * NSA_H + h;

  if (L < 16) {  // gate branch 0 for the 16 query rows
    const float* qp = q + ((size_t)(bi * NSA_SEQ + q0 + L) * NSA_QH + hoff) * NSA_DIM;
    float g = bg[0];
    for (int e = 0; e < NSA_DIM; ++e) g += qp[e] * wg[e * 3 + 0];
    ldsG[L] = 1.0f / (1.0f + expf(-g));
  }

  v2f qa[32];
  {
    const float* qp = q + ((size_t)(bi * NSA_SEQ + q0 + lo) * NSA_QH + hoff) * NSA_DIM;
#pragma unroll
    for (int kk = 0; kk < 32; ++kk) qa[kk] = *(const v2f*)(qp + 4 * kk + 2 * hi);
  }
  const float* kcb = kc + (size_t)((bi * NSA_KV + kvi) * NSA_NBLK) * NSA_DIM;
  const float* vcb = vc + (size_t)((bi * NSA_KV + kvi) * NSA_NBLK) * NSA_DIM;

  v8f s0 = qk_tile(qa, kcb + lo * NSA_DIM, hi);
  v8f s1 = qk_tile(qa, kcb + (16 + lo) * NSA_DIM, hi);

  float lrow[8];
  v8f p0, p1;
#pragma unroll
  for (int r = 0; r < 8; ++r) {
    float a = s0[r] * NSA_SCALE, b2 = s1[r] * NSA_SCALE;
    float m = redmax16(fmaxf(a, b2));
    p0[r] = expf(a - m);
    p1[r] = expf(b2 - m);
    lrow[r] = redsum16(p0[r] + p1[r]);
  }
#pragma unroll
  for (int r = 0; r < 8; ++r) {
    ldsP[(r + 8 * hi) * 32 + lo] = p0[r];
    ldsP[(r + 8 * hi) * 32 + 16 + lo] = p1[r];
  }
  __syncthreads();

  {  // pooled[n] += sum over rows of normalized P
    float c0 = 0.0f, c1 = 0.0f;
#pragma unroll
    for (int r = 0; r < 8; ++r) {
      float inv = 1.0f / lrow[r];
      c0 += p0[r] * inv;
      c1 += p1[r] * inv;
    }
    c0 += __shfl_xor(c0, 16, 32);
    c1 += __shfl_xor(c1, 16, 32);
    if (hi == 0) {
      float* pr = pooled + (size_t)((bi * NSA_KV + kvi) * NSA_NBLK + (qt >> 2)) * NSA_NBLK;
      atomicAdd(pr + lo, c0);
      atomicAdd(pr + 16 + lo, c1);
    }
  }

  v8f zero = {};
  v8f o[8];
#pragma unroll
  for (int dt = 0; dt < 8; ++dt) o[dt] = zero;
#pragma unroll
  for (int dt = 0; dt < 8; ++dt) {
#pragma unroll
    for (int kk = 0; kk < 8; ++kk) {  // 32 keys, K-chunks of 4
      v2f a = *(const v2f*)(&ldsP[lo * 32 + 4 * kk + 2 * hi]);
      int j = 4 * kk + 2 * hi;
      v2f b;
      b.x = vcb[j * NSA_DIM + dt * 16 + lo];
      b.y = vcb[(j + 1) * NSA_DIM + dt * 16 + lo];
      o[dt] = wmma_f32(a, b, o[dt]);
    }
  }
#pragma unroll
  for (int dt = 0; dt < 8; ++dt)
#pragma unroll
    for (int r = 0; r < 8; ++r) {
      int row = r + 8 * hi;
      size_t oi = ((size_t)(bi * NSA_SEQ + q0 + row) * NSA_QH + hoff) * NSA_DIM + dt * 16 + lo;
      out[oi] += ldsG[row] * o[dt][r] / lrow[r];
    }
}

// ---------------- kernel 3: top-16 of 32 per (b,kv,qblock) ----------------
__global__ void nsa_topk(const float* __restrict__ pooled, int* __restrict__ idx) {
  int t = threadIdx.x;
  if (t >= 128) return;
  float vals[32];
  for (int n = 0; n < 32; ++n) vals[n] = pooled[t * 32 + n];
  for (int s = 0; s < 16; ++s) {
    int best = 0;
    float bv = vals[0];
    for (int n = 1; n < 32; ++n)
      if (vals[n] > bv) { bv = vals[n]; best = n; }
    idx[t * 16 + s] = best;
    vals[best] = -3.4e38f;
  }
}

// ---------------- kernel 4: selected-block attention ----------------
__global__ __launch_bounds__(32) void nsa_slc_attn(
    const float* __restrict__ q, const float* __restrict__ k,
    const float* __restrict__ v, const int* __restrict__ idx,
    const float* __restrict__ wg, const float* __restrict__ bg,
    float* __restrict__ out) {
  __shared__ float ldsP[256];
  __shared__ float ldsG[16];
  __shared__ int ldsI[16];
  const int w = blockIdx.x;
  const int sub = w & 3, h = (w >> 2) & 3, nqb = (w >> 4) & 31;
  const int kvi = (w >> 9) & 1, bi = w >> 10;
  const int L = threadIdx.x, lo = L & 15, hi = L >> 4;
  const int q0 = nqb * 64 + sub * 16;
  const int hoff = kvi * NSA_H + h;

  if (L < 16) {
    ldsI[L] = idx[((bi * NSA_KV + kvi) * NSA_NBLK + nqb) * 16 + L];
    const float* qp = q + ((size_t)(bi * NSA_SEQ + q0 + L) * NSA_QH + hoff) * NSA_DIM;
    float g = bg[1];
    for (int e = 0; e < NSA_DIM; ++e) g += qp[e] * wg[e * 3 + 1];
    ldsG[L] = 1.0f / (1.0f + expf(-g));
  }
  v2f qa[32];
  {
    const float* qp = q + ((size_t)(bi * NSA_SEQ + q0 + lo) * NSA_QH + hoff) * NSA_DIM;
#pragma unroll
    for (int kk = 0; kk < 32; ++kk) qa[kk] = *(const v2f*)(qp + 4 * kk + 2 * hi);
  }
  __syncthreads();

  float mrow[8], lrow[8];
#pragma unroll
  for (int r = 0; r < 8; ++r) { mrow[r] = -1e30f; lrow[r] = 0.0f; }
  v8f zero = {};
  v8f o[8];
#pragma unroll
  for (int dt = 0; dt < 8; ++dt) o[dt] = zero;

  for (int t = 0; t < 16; ++t) {
    const int blk = ldsI[t];
    const float* kb = k + (size_t)(bi * NSA_SEQ + blk * 64) * KV_STRIDE + kvi * NSA_DIM;
    const float* vb = v + (size_t)(bi * NSA_SEQ + blk * 64) * KV_STRIDE + kvi * NSA_DIM;
    for (int c = 0; c < 4; ++c) {
      v8f s = qk_tile(qa, kb + (size_t)(c * 16 + lo) * KV_STRIDE, hi);
      v8f p;
#pragma unroll
      for (int r = 0; r < 8; ++r) {
        float sv = s[r] * NSA_SCALE;
        float mn = fmaxf(mrow[r], redmax16(sv));
        float corr = expf(mrow[r] - mn);
        float pv = expf(sv - mn);
        lrow[r] = lrow[r] * corr + redsum16(pv);
        mrow[r] = mn;
        p[r] = pv;
#pragma unroll
        for (int dt = 0; dt < 8; ++dt) o[dt][r] *= corr;
      }
      __syncthreads();
#pragma unroll
      for (int r = 0; r < 8; ++r) ldsP[(r + 8 * hi) * 16 + lo] = p[r];
      __syncthreads();
#pragma unroll
      for (int dt = 0; dt < 8; ++dt) {
#pragma unroll
        for (int kk = 0; kk < 4; ++kk) {
          v2f a = *(const v2f*)(&ldsP[lo * 16 + 4 * kk + 2 * hi]);
          int j = c * 16 + 4 * kk + 2 * hi;
          v2f b;
          b.x = vb[(size_t)j * KV_STRIDE + dt * 16 + lo];
          b.y = vb[(size_t)(j + 1) * KV_STRIDE + dt * 16 + lo];
          o[dt] = wmma_f32(a, b, o[dt]);
        }
      }
    }
  }
#pragma unroll
  for (int dt = 0; dt < 8; ++dt)
#pragma unroll
    for (int r = 0; r < 8; ++r) {
      int row = r + 8 * hi;
      size_t oi = ((size_t)(bi * NSA_SEQ + q0 + row) * NSA_QH + hoff) * NSA_DIM + dt * 16 + lo;
      out[oi] += ldsG[row] * o[dt][r] / lrow[r];
    }
}

// ---------------- kernel 5: sliding-window attention ----------------
__global__ __launch_bounds__(32) void nsa_win_attn(
    const float* __restrict__ q, const float* __restrict__ k,
    const float* __restrict__ v, const float* __restrict__ wg,
    const float* __restrict__ bg, float* __restrict__ out) {
  __shared__ float ldsP[256];
  __shared__ float ldsG[16];
  const int w = blockIdx.x;
  const int qt = w & 127, h = (w >> 7) & 3, kvi = (w >> 9) & 1, bi = w >> 10;
  const int L = threadIdx.x, lo = L & 15, hi = L >> 4;
  const int i0 = qt * 16;
  const int hoff = kvi * NSA_H + h;

  if (L < 16) {
    const float* qp = q + ((size_t)(bi * NSA_SEQ + i0 + L) * NSA_QH + hoff) * NSA_DIM;
    float g = bg[2];
    for (int e = 0; e < NSA_DIM; ++e) g += qp[e] * wg[e * 3 + 2];
    ldsG[L] = 1.0f / (1.0f + expf(-g));
  }
  v2f qa[32];
  {
    const float* qp = q + ((size_t)(bi * NSA_SEQ + i0 + lo) * NSA_QH + hoff) * NSA_DIM;
#pragma unroll
    for (int kk = 0; kk < 32; ++kk) qa[kk] = *(const v2f*)(qp + 4 * kk + 2 * hi);
  }
  __syncthreads();

  float mrow[8], lrow[8];
#pragma unroll
  for (int r = 0; r < 8; ++r) { mrow[r] = -1e30f; lrow[r] = 0.0f; }
  v8f zero = {};
  v8f o[8];
#pragma unroll
  for (int dt = 0; dt < 8; ++dt) o[dt] = zero;

  int jstart = i0 - 512;
  if (jstart < 0) jstart = 0;
  for (int jt = jstart; jt <= i0; jt += 16) {
    const float* kb = k + (size_t)(bi * NSA_SEQ + jt) * KV_STRIDE + kvi * NSA_DIM;
    const float* vb = v + (size_t)(bi * NSA_SEQ + jt) * KV_STRIDE + kvi * NSA_DIM;
    v8f s = qk_tile(qa, kb + (size_t)lo * KV_STRIDE, hi);
    v8f p;
#pragma unroll
    for (int r = 0; r < 8; ++r) {
      int irow = i0 + r + 8 * hi;
      int j = jt + lo;
      bool valid = (j <= irow) && (j >= irow - 512);
      float sv = s[r] * NSA_SCALE;
      float sm = valid ? sv : -1e30f;
      float mn = fmaxf(mrow[r], redmax16(sm));
      float corr = expf(mrow[r] - mn);
      float pv = valid ? expf(sv - mn) : 0.0f;
      lrow[r] = lrow[r] * corr + redsum16(pv);
      mrow[r] = mn;
      p[r] = pv;
#pragma unroll
      for (int dt = 0; dt < 8; ++dt) o[dt][r] *= corr;
    }
    __syncthreads();
#pragma unroll
    for (int r = 0; r < 8; ++r) ldsP[(r + 8 * hi) * 16 + lo] = p[r];
    __syncthreads();
#pragma unroll
    for (int dt = 0; dt < 8; ++dt) {
#pragma unroll
      for (int kk = 0; kk < 4; ++kk) {
        v2f a = *(const v2f*)(&ldsP[lo * 16 + 4 * kk + 2 * hi]);
        int j = 4 * kk + 2 * hi;
        v2f b;
        b.x = vb[(size_t)j * KV_STRIDE + dt * 16 + lo];
        b.y = vb[(size_t)(j + 1) * KV_STRIDE + dt * 16 + lo];
        o[dt] = wmma_f32(a, b, o[dt]);
      }
    }
  }
#pragma unroll
  for (int dt = 0; dt < 8; ++dt)
#pragma unroll
    for (int r = 0; r < 8; ++r) {
      int row = r + 8 * hi;
      size_t oi = ((size_t)(bi * NSA_SEQ + i0 + row) * NSA_QH + hoff) * NSA_DIM + dt * 16 + lo;
      out[oi] += ldsG[row] * o[dt][r] / lrow[r];
    }
}

extern "C" void kernel_launch(void* const* d_in, const int* in_sizes, int n_in,
                              void* d_out, int out_size, void* d_ws, size_t ws_size,
                              hipStream_t stream) {
  const float* q      = (const float*)d_in[0];
  const float* k      = (const float*)d_in[1];
  const float* v      = (const float*)d_in[2];
  const float* w_k    = (const float*)d_in[3];
  const float* b_k    = (const float*)d_in[4];
  const float* w_v    = (const float*)d_in[5];
  const float* b_v    = (const float*)d_in[6];
  const float* w_gate = (const float*)d_in[7];
  const float* b_gate = (const float*)d_in[8];
  float* out = (float*)d_out;

  float* kc     = (float*)d_ws;        // 2*2*32*128 = 16384 floats
  float* vc     = kc + 16384;          // 16384 floats
  float* pooled = vc + 16384;          // 2*2*32*32 = 4096 floats
  int*   idx    = (int*)(pooled + 4096);  // 2*2*32*16 = 2048 ints

  nsa_zero<<<(out_size + 255) / 256, 256, 0, stream>>>(out, out_size);
  nsa_compress<<<128, 256, 0, stream>>>(k, v, w_k, b_k, w_v, b_v, kc, vc, pooled);
  nsa_cmp_attn<<<2048, 32, 0, stream>>>(q, kc, vc, w_gate, b_gate, pooled, out);
  nsa_topk<<<1, 128, 0, stream>>>(pooled, idx);
  nsa_slc_attn<<<2048, 32, 0, stream>>>(q, k, v, idx, w_gate, b_gate, out);
  nsa_win_attn<<<2048, 32, 0, stream>>>(q, k, v, w_gate, b_gate, out);
}